// WaveletLayer_81750407512695
// MI455X (gfx1250) — compile-verified
//
#include <hip/hip_runtime.h>
#include <hip/hip_bf16.h>
#include <math.h>

// ---------------------------------------------------------------------------
// WaveletLayer for MI455X (gfx1250), fp32 path via V_WMMA_F32_16X16X4_F32.
//
//   h  = conv1d_k13(x, Win) + bin                     (B=8,H=128,L=8192)
//   out_even[l] = Ceff(3-tap) * h, out_odd[l] = Deff(2-tap) * h  (interleaved)
//   + one-column fixup for env[15] boundary term.
//
// v2: each wave computes 2 M-tiles x 4 N-tiles (8 accumulators) so every
//     B-fragment ds_load_b64 feeds 2 WMMAs (halved LDS traffic per FLOP);
//     workgroup tile widened to 128 positions; A-row prefetch added.
//
// Workspace layout (floats):
//   [0)            Wt      : 128*832      (k-major transposed conv_in_w)
//   [WS_H)         h       : 8*128*8192
//   [WS_CEFF)      Ceff    : 64*384
//   [WS_DEFF)      Deff    : 64*256
//   [WS_AUX)       aux     : 128*2        (e1, env15 per channel)
// total ~34.2 MB
// ---------------------------------------------------------------------------

#define L_IN   8192
#define CIN    64
#define HCH    128
#define KW     13
#define KTOT   (CIN * KW)      // 832
#define COUT   64
#define LOUT   (2 * L_IN)      // 16384
#define NTILE  128             // output positions per workgroup
#define NBATCH 8

#define XS_STRIDE 66           // 64 + 2 pad (bank-conflict avoidance)
#define HS_STRIDE 130          // 128 + 2 pad
#define XS_POS    140          // 128 + 12 halo
#define HS_POS    130          // 128 + 2 halo

typedef float v2f __attribute__((ext_vector_type(2)));
typedef float v8f __attribute__((ext_vector_type(8)));

__device__ __forceinline__ v8f wmma4(v2f a, v2f b, v8f c) {
  // D = A(16x4 f32) * B(4x16 f32) + C(16x16 f32)
  return __builtin_amdgcn_wmma_f32_16x16x4_f32(
      /*neg_a=*/false, a, /*neg_b=*/false, b,
      /*c_mod=*/(short)0, c, /*reuse_a=*/false, /*reuse_b=*/false);
}

// --------------------------- prep kernels ----------------------------------

// Transpose conv_in_w (H, Cin, 13) -> Wt (H, 832) with k = t*64 + ic
__global__ __launch_bounds__(256) void prep_wt(const float* __restrict__ w_in,
                                               float* __restrict__ wt) {
  int i = blockIdx.x * blockDim.x + threadIdx.x;
  if (i >= HCH * KTOT) return;
  int hch = i / KTOT;
  int k   = i % KTOT;
  int t   = k >> 6;
  int ic  = k & 63;
  wt[i] = w_in[(hch * CIN + ic) * KW + t];
}

// Build effective conv_out weights from env sums, plus aux (e1, env15).
__global__ __launch_bounds__(256) void prep_eff(const float* __restrict__ w_out,
                                                const float* __restrict__ wscale,
                                                float* __restrict__ ceff,
                                                float* __restrict__ deff,
                                                float* __restrict__ aux) {
  int i = blockIdx.x * blockDim.x + threadIdx.x;
  if (i >= COUT * HCH) return;
  int oc = i / HCH;
  int ic = i % HCH;
  float scale = wscale[ic];
  float e0 = 0.f, e1 = 0.f, env15 = 0.f;
  #pragma unroll
  for (int j = 0; j < 16; ++j) {
    float sp  = -10.f + (20.f / 15.f) * (float)j;
    float env = cosf(sp) * expf(-sp * sp / scale);
    if (j < 8) e0 += env; else e1 += env;
    if (j == 15) env15 = env;
  }
  float E = e0 + e1;
  const float* w = w_out + (size_t)(oc * HCH + ic) * 3;
  float w0 = w[0], w1 = w[1], w2 = w[2];
  // even outputs (m = 2l): taps over h[l-1], h[l], h[l+1]
  ceff[(size_t)oc * 384 + 0 * HCH + ic] = e1 * w0;
  ceff[(size_t)oc * 384 + 1 * HCH + ic] = e0 * w0 + E * w1 + e1 * w2;
  ceff[(size_t)oc * 384 + 2 * HCH + ic] = e0 * w2;
  // odd outputs (m = 2l+1): taps over h[l], h[l+1]
  deff[(size_t)oc * 256 + 0 * HCH + ic] = E * w0 + e1 * w1;
  deff[(size_t)oc * 256 + 1 * HCH + ic] = e0 * w1 + E * w2;
  if (oc == 0) { aux[ic * 2] = e1; aux[ic * 2 + 1] = env15; }
}

// --------------------------- conv_in (WMMA) --------------------------------
// Grid: B * (L/128) blocks, 256 threads (8 waves).
// Wave w: M-pair (w&3)*32 (2 tiles of 16 rows), N-half (w>>2)*64 (4 tiles).
// Each K-step: 2 A global loads, 4 B ds loads, 8 WMMAs (B reused 2x).
__global__ __launch_bounds__(256) void conv_in_wmma(const float* __restrict__ x,
                                                    const float* __restrict__ wt,
                                                    const float* __restrict__ bias,
                                                    float* __restrict__ h) {
  __shared__ float xs[XS_POS * XS_STRIDE];       // positions l0-6 .. l0+133
  const int b    = blockIdx.x >> 6;              // / 64 tiles
  const int tile = blockIdx.x & 63;
  const int l0   = tile * NTILE;
  const int tid  = threadIdx.x;

  {// stage x patch into LDS as xs[p][ic] (ic contiguous, padded stride)
    const int ic = tid >> 2;
    const float* xrow = x + ((size_t)b * CIN + ic) * L_IN;
    for (int p = (tid & 3); p < XS_POS; p += 4) {
      int l = l0 - 6 + p;
      xs[p * XS_STRIDE + ic] = (l >= 0 && l < L_IN) ? xrow[l] : 0.0f;
    }
  }
  __syncthreads();

  const int wave    = tid >> 5;
  const int lane    = tid & 31;
  const int laneRow = lane & 15;
  const int hi      = lane >> 4;
  const int koff    = hi * 2;                    // K pair (0,1) or (2,3)
  const int m0      = (wave & 3) * 32;           // 2 M-tiles: m0, m0+16
  const int nbase   = (wave >> 2) * 64;          // 4 N-tiles

  v8f acc[2][4] = {};
  const float* arow0 = wt + (size_t)(m0 + laneRow) * KTOT;
  const float* arow1 = wt + (size_t)(m0 + 16 + laneRow) * KTOT;
  __builtin_prefetch(arow0, 0, 1);               // global_prefetch_b8
  __builtin_prefetch(arow1, 0, 1);

  #pragma unroll 2
  for (int k0 = 0; k0 < KTOT; k0 += 4) {
    int k = k0 + koff;                           // even, 8B-aligned pairs
    v2f a0 = *(const v2f*)(arow0 + k);
    v2f a1 = *(const v2f*)(arow1 + k);
    int t  = k >> 6;                             // tap
    int ic = k & 63;
    const float* bs = &xs[t * XS_STRIDE + ic];
    #pragma unroll
    for (int nt = 0; nt < 4; ++nt) {
      v2f bv = *(const v2f*)(bs + (nbase + nt * 16 + laneRow) * XS_STRIDE);
      acc[0][nt] = wmma4(a0, bv, acc[0][nt]);
      acc[1][nt] = wmma4(a1, bv, acc[1][nt]);
    }
  }

  // write back: C layout -> VGPR j, lanes<16: row j, col lane; lanes>=16: row j+8
  float* hb = h + (size_t)b * HCH * L_IN;
  const int col = lane & 15;
  #pragma unroll
  for (int mt = 0; mt < 2; ++mt) {
    #pragma unroll
    for (int nt = 0; nt < 4; ++nt) {
      #pragma unroll
      for (int j = 0; j < 8; ++j) {
        int ch = m0 + mt * 16 + j + 8 * hi;
        hb[(size_t)ch * L_IN + l0 + nbase + nt * 16 + col] = acc[mt][nt][j] + bias[ch];
      }
    }
  }
}

// --------------------------- conv_out (WMMA) -------------------------------
// Waves 0-3: even-parity 3-tap GEMM (K=384); waves 4-7: odd-parity 2-tap
// GEMM (K=256). Within each parity: wave takes M-pair (w&1)*32 and N-half
// ((w>>1)&1)*64. Stores interleave with stride 2 into out.
__global__ __launch_bounds__(256) void conv_out_wmma(const float* __restrict__ h,
                                                     const float* __restrict__ ceff,
                                                     const float* __restrict__ deff,
                                                     const float* __restrict__ bias2,
                                                     float* __restrict__ out) {
  __shared__ float hs[HS_POS * HS_STRIDE];       // positions l0-1 .. l0+128
  const int b    = blockIdx.x >> 6;
  const int tile = blockIdx.x & 63;
  const int l0   = tile * NTILE;
  const int tid  = threadIdx.x;

  {// stage h patch into LDS as hs[p][ic]
    const int ic = tid >> 1;
    const float* hrow = h + ((size_t)b * HCH + ic) * L_IN;
    for (int p = (tid & 1); p < HS_POS; p += 2) {
      int l = l0 - 1 + p;
      hs[p * HS_STRIDE + ic] = (l >= 0 && l < L_IN) ? hrow[l] : 0.0f;
    }
  }
  __syncthreads();

  const int wave    = tid >> 5;
  const int lane    = tid & 31;
  const int laneRow = lane & 15;
  const int hi      = lane >> 4;
  const int koff    = hi * 2;
  const bool odd    = wave >= 4;
  const int  pw     = wave & 3;                  // wave index within parity
  const int  m0     = (pw & 1) * 32;             // 2 M-tiles: m0, m0+16
  const int  nbase  = (pw >> 1) * 64;            // 4 N-tiles
  const int  KT     = odd ? 256 : 384;
  const int  pbase  = odd ? 1 : 0;               // odd taps start at h[l]
  const float* A    = odd ? deff : ceff;
  const float* arow0 = A + (size_t)(m0 + laneRow) * KT;
  const float* arow1 = A + (size_t)(m0 + 16 + laneRow) * KT;
  __builtin_prefetch(arow0, 0, 1);
  __builtin_prefetch(arow1, 0, 1);

  v8f acc[2][4] = {};
  for (int k0 = 0; k0 < KT; k0 += 4) {
    int k = k0 + koff;
    v2f a0 = *(const v2f*)(arow0 + k);
    v2f a1 = *(const v2f*)(arow1 + k);
    int tap = k >> 7;
    int ic  = k & 127;
    const float* bs = &hs[(tap + pbase) * HS_STRIDE + ic];
    #pragma unroll
    for (int nt = 0; nt < 4; ++nt) {
      v2f bv = *(const v2f*)(bs + (nbase + nt * 16 + laneRow) * HS_STRIDE);
      acc[0][nt] = wmma4(a0, bv, acc[0][nt]);
      acc[1][nt] = wmma4(a1, bv, acc[1][nt]);
    }
  }

  const int col    = lane & 15;
  const int parity = odd ? 1 : 0;
  float* ob = out + (size_t)b * COUT * LOUT;
  #pragma unroll
  for (int mt = 0; mt < 2; ++mt) {
    #pragma unroll
    for (int nt = 0; nt < 4; ++nt) {
      #pragma unroll
      for (int j = 0; j < 8; ++j) {
        int oc = m0 + mt * 16 + j + 8 * hi;
        int l  = l0 + nbase + nt * 16 + col;
        ob[(size_t)oc * LOUT + 2 * l + parity] = acc[mt][nt][j] + bias2[oc];
      }
    }
  }
}

// ---- exact boundary fixup: last odd column uses env[15] instead of e1 -----
__global__ __launch_bounds__(256) void fixup_last(const float* __restrict__ h,
                                                  const float* __restrict__ w_out,
                                                  const float* __restrict__ aux,
                                                  float* __restrict__ out) {
  int i = blockIdx.x * blockDim.x + threadIdx.x;
  if (i >= NBATCH * COUT) return;
  int b  = i / COUT;
  int oc = i % COUT;
  float s = 0.f;
  for (int ic = 0; ic < HCH; ++ic) {
    float w1 = w_out[(size_t)(oc * HCH + ic) * 3 + 1];
    s += w1 * (aux[ic * 2 + 1] - aux[ic * 2]) *
         h[((size_t)b * HCH + ic) * L_IN + (L_IN - 1)];
  }
  out[((size_t)b * COUT + oc) * LOUT + (LOUT - 1)] += s;
}

// ------------------------------- launcher ----------------------------------
extern "C" void kernel_launch(void* const* d_in, const int* in_sizes, int n_in,
                              void* d_out, int out_size, void* d_ws, size_t ws_size,
                              hipStream_t stream) {
  const float* x       = (const float*)d_in[0];
  const float* w_in    = (const float*)d_in[1];
  const float* b_in    = (const float*)d_in[2];
  const float* w_out   = (const float*)d_in[3];
  const float* b_out   = (const float*)d_in[4];
  const float* wscale  = (const float*)d_in[5];
  float*       out     = (float*)d_out;
  float*       ws      = (float*)d_ws;

  // workspace offsets (floats)
  const size_t OFF_WT   = 0;
  const size_t OFF_H    = OFF_WT + (size_t)HCH * KTOT;            // 106496
  const size_t OFF_CEFF = OFF_H + (size_t)NBATCH * HCH * L_IN;    // + 8388608
  const size_t OFF_DEFF = OFF_CEFF + (size_t)COUT * 384;
  const size_t OFF_AUX  = OFF_DEFF + (size_t)COUT * 256;

  float* wt   = ws + OFF_WT;
  float* h    = ws + OFF_H;
  float* ceff = ws + OFF_CEFF;
  float* deff = ws + OFF_DEFF;
  float* aux  = ws + OFF_AUX;

  prep_wt <<<(HCH * KTOT + 255) / 256, 256, 0, stream>>>(w_in, wt);
  prep_eff<<<(COUT * HCH + 255) / 256, 256, 0, stream>>>(w_out, wscale, ceff, deff, aux);

  conv_in_wmma <<<NBATCH * (L_IN / NTILE), 256, 0, stream>>>(x, wt, b_in, h);
  conv_out_wmma<<<NBATCH * (L_IN / NTILE), 256, 0, stream>>>(h, ceff, deff, b_out, out);
  fixup_last   <<<(NBATCH * COUT + 255) / 256, 256, 0, stream>>>(h, w_out, aux, out);
}